// GraphSequenceOrderer_53257594470659
// MI455X (gfx1250) — compile-verified
//
#include <hip/hip_runtime.h>
#include <stdint.h>

#define BATCH 32
#define KSLOT 1024
#define DDIM  256

// ---------------------------------------------------------------------------
// Kernel 1: Hilbert index + stable bitonic argsort per batch (1 block/batch).
// key = (hilbert_d << 10) | k  ->  ascending sort == stable argsort of d.
// ---------------------------------------------------------------------------
__global__ __launch_bounds__(1024) void hilbert_sort_kernel(
    const float* __restrict__ centroids,   // [B, K, 2]
    float* __restrict__ out_order_f,       // [B, K] (float-cast ints)
    float* __restrict__ out_rev_f,         // [B, K]
    int* __restrict__ ws_order)            // [B, K] int scratch for gather
{
    __shared__ uint32_t key[KSLOT];
    const int b = blockIdx.x;
    const int t = threadIdx.x;

    const size_t cbase = ((size_t)b * KSLOT + t) * 2;
    float fx = centroids[cbase + 0] * 15.0f;
    float fy = centroids[cbase + 1] * 15.0f;
    int x = (int)fminf(fmaxf(fx, 0.0f), 15.0f);   // clip then trunc, as jnp
    int y = (int)fminf(fmaxf(fy, 0.0f), 15.0f);

    int d = 0;
#pragma unroll
    for (int s = 8; s > 0; s >>= 1) {
        int rx = (x & s) ? 1 : 0;
        int ry = (y & s) ? 1 : 0;
        d += s * s * ((3 * rx) ^ ry);
        if (ry == 0) {                 // swap quadrant
            int ox = x, oy = y;
            if (rx) { x = s - 1 - oy; y = s - 1 - ox; }  // + reflect
            else    { x = oy;         y = ox;         }
        }
    }

    key[t] = ((uint32_t)d << 10) | (uint32_t)t;

    // bitonic sort of 1024 keys, ascending
    for (int kk = 2; kk <= KSLOT; kk <<= 1) {
        for (int j = kk >> 1; j > 0; j >>= 1) {
            __syncthreads();
            int ixj = t ^ j;
            if (ixj > t) {
                uint32_t a = key[t], c = key[ixj];
                bool asc = ((t & kk) == 0);
                if ((a > c) == asc) { key[t] = c; key[ixj] = a; }
            }
        }
    }
    __syncthreads();

    int ord = (int)(key[t] & 1023u);
    ws_order[b * KSLOT + t]    = ord;
    out_order_f[b * KSLOT + t] = (float)ord;
    out_rev_f[b * KSLOT + ord] = (float)t;   // inverse permutation scatter
}

// ---------------------------------------------------------------------------
// Kernel 2: indirect row gather via CDNA5 async global<->LDS copy engine.
// 8 waves/block, 2 rows (2 KiB) per wave: 4 async B128 loads in flight
// (512 B each) into per-wave LDS slabs, one s_wait_asynccnt, then 4 async
// stores out. Final wait omitted: s_endpgm does an implicit s_wait_idle.
// ---------------------------------------------------------------------------
__global__ __launch_bounds__(256) void gather_async_kernel(
    const float* __restrict__ slots,     // [B, K, D]
    const int* __restrict__ ws_order,    // [B*K]
    float* __restrict__ out)             // [B, K, D]
{
    __shared__ float smem[16 * DDIM];    // 16 KiB: 2 KiB slab per wave
    const int lane = threadIdx.x & 31;
    const int wave = threadIdx.x >> 5;
    const int row0 = blockIdx.x * 16 + wave * 2;   // flattened b*K + k
    const int row1 = row0 + 1;
    const int b0   = row0 >> 10;
    const int b1   = row1 >> 10;
    const int s0   = ws_order[row0];
    const int s1   = ws_order[row1];

    const uint64_t lane_off = (uint64_t)(lane * 16);
    uint64_t ga0 = (uint64_t)(uintptr_t)(slots + ((size_t)b0 * KSLOT + s0) * DDIM) + lane_off;
    uint64_t ga1 = (uint64_t)(uintptr_t)(slots + ((size_t)b1 * KSLOT + s1) * DDIM) + lane_off;
    uint64_t gd0 = (uint64_t)(uintptr_t)(out + (size_t)row0 * DDIM) + lane_off;
    uint64_t gd1 = gd0 + 1024;           // dst rows are contiguous

    // low 32 bits of generic LDS pointer == workgroup-relative LDS byte offset
    uint32_t lds0 = (uint32_t)(uintptr_t)(&smem[0]) +
                    (uint32_t)(wave * 2048) + (uint32_t)(lane * 16);
    uint32_t lds1 = lds0 + 1024;

    asm volatile(
        "global_load_async_to_lds_b128 %0, %2, off\n\t"
        "global_load_async_to_lds_b128 %0, %2, off offset:512\n\t"
        "global_load_async_to_lds_b128 %1, %3, off\n\t"
        "global_load_async_to_lds_b128 %1, %3, off offset:512\n\t"
        "s_wait_asynccnt 0\n\t"
        "global_store_async_from_lds_b128 %4, %0, off\n\t"
        "global_store_async_from_lds_b128 %4, %0, off offset:512\n\t"
        "global_store_async_from_lds_b128 %5, %1, off\n\t"
        "global_store_async_from_lds_b128 %5, %1, off offset:512"
        :
        : "v"(lds0), "v"(lds1), "v"(ga0), "v"(ga1), "v"(gd0), "v"(gd1)
        : "memory");
}

extern "C" void kernel_launch(void* const* d_in, const int* in_sizes, int n_in,
                              void* d_out, int out_size, void* d_ws, size_t ws_size,
                              hipStream_t stream) {
    const float* slots     = (const float*)d_in[0];
    // d_in[1] = adj : unused by the reference output -> never touched (saves 128 MiB of traffic)
    const float* centroids = (const float*)d_in[2];

    float* out        = (float*)d_out;
    float* out_order  = out + (size_t)BATCH * KSLOT * DDIM;
    float* out_rev    = out_order + (size_t)BATCH * KSLOT;
    int*   ws_order   = (int*)d_ws;

    hipLaunchKernelGGL(hilbert_sort_kernel, dim3(BATCH), dim3(KSLOT), 0, stream,
                       centroids, out_order, out_rev, ws_order);

    hipLaunchKernelGGL(gather_async_kernel, dim3(BATCH * KSLOT / 16), dim3(256), 0, stream,
                       slots, ws_order, out);
}